// LearnableFeatureDroppingBySpectral_9783935500490
// MI455X (gfx1250) — compile-verified
//
#include <hip/hip_runtime.h>
#include <hip/hip_bf16.h>
#include <math.h>

#define N_NODES  20000
#define N_FEAT   128
#define EIGD     32
#define HID      64
#define M_TOT    (N_NODES * N_FEAT)   /* 2,560,000 rows */
#define N_EDGES  640000
#define NUM_DROP (M_TOT / 10)         /* 256,000 */
#define NBINS1   131072               /* top 17 bits of float */
#define NBINS2   32768                /* low 15 bits */
#define EPS_LOG  1e-8f
#define BSTRIDE  80                   /* float2 stride per (ch,half) block: 160 dwords = 32 mod 64 banks */

typedef __attribute__((ext_vector_type(2))) float v2f;
typedef __attribute__((ext_vector_type(8))) float v8f;

// ---------------------------------------------------------------- zero ws
__global__ void __launch_bounds__(256) zero_ws_kernel(unsigned int* p, int n) {
  int i = blockIdx.x * blockDim.x + threadIdx.x;
  if (i < n) p[i] = 0u;
}

// ------------------------------------------- fused GEMM + gumbel + hist1
// One wave handles a 16-row tile of flat[M, 32].
// h = relu(tile @ W1 + b1) via V_WMMA_F32_16X16X4_F32 (8 K-chunks x 4 N-tiles).
// B fragments pre-packed in LDS as float2 -> single ds_load_b64 per fragment,
// halves bank-disjoint via BSTRIDE padding.
__global__ void __launch_bounds__(256)
gemm_poss_kernel(const float* __restrict__ eig,   // [M, 32]
                 const float* __restrict__ W1,    // [32, 64]
                 const float* __restrict__ b1,    // [64]
                 const float* __restrict__ W2,    // [64, 2]
                 const float* __restrict__ b2,    // [2]
                 const float* __restrict__ uni,   // [M, 2]
                 float* __restrict__ poss,        // [M]
                 unsigned int* __restrict__ hist1)
{
  __shared__ v2f   sW1p[16 * BSTRIDE];  // packed (kb,kb+1) pairs, 10.24 KB
  __shared__ float sb1[HID];
  __shared__ float sW2[HID * 2];

  const int tid = threadIdx.x;
  // pack W1: block ch2 = ch*2+half covers K rows (kb, kb+1), column n
  for (int t = tid; t < 16 * HID; t += 256) {
    const int ch2 = t >> 6;
    const int n   = t & 63;
    const int kb  = (ch2 >> 1) * 4 + (ch2 & 1) * 2;
    v2f p;
    p.x = W1[kb * HID + n];
    p.y = W1[(kb + 1) * HID + n];
    sW1p[ch2 * BSTRIDE + n] = p;
  }
  if (tid < HID)     sb1[tid] = b1[tid];
  if (tid < HID * 2) sW2[tid] = W2[tid];
  __syncthreads();

  const int wave = tid >> 5;
  const int lane = tid & 31;
  const int half = lane >> 4;        // 0: lanes 0-15, 1: lanes 16-31
  const int mrow = lane & 15;

  const long long r0 = ((long long)blockIdx.x * 8 + wave) * 16;

  // A fragments: 16x4 f32 layout -> lane<16 holds K = kb,kb+1; lane>=16 K = kb+2,kb+3
  const float* arow = eig + (r0 + mrow) * EIGD + half * 2;
  v2f a[8];
#pragma unroll
  for (int c = 0; c < 8; ++c)
    a[c] = *(const v2f*)(arow + c * 4);

  const float b2v0 = b2[0];
  const float b2v1 = b2[1];

  float p0[8], p1[8];
#pragma unroll
  for (int v = 0; v < 8; ++v) { p0[v] = 0.f; p1[v] = 0.f; }

#pragma unroll
  for (int j = 0; j < 4; ++j) {            // HID = 4 x 16 column tiles
    v8f c = {};
#pragma unroll
    for (int ch = 0; ch < 8; ++ch) {       // K = 8 x 4
      const v2f b = sW1p[(ch * 2 + half) * BSTRIDE + j * 16 + mrow];
      c = __builtin_amdgcn_wmma_f32_16x16x4_f32(false, a[ch], false, b,
                                                (short)0, c, false, false);
    }
    // c[v] holds (M = v + half*8, N = j*16 + mrow)
    const float bias = sb1[j * 16 + mrow];
    const float w2c0 = sW2[(j * 16 + mrow) * 2 + 0];
    const float w2c1 = sW2[(j * 16 + mrow) * 2 + 1];
#pragma unroll
    for (int v = 0; v < 8; ++v) {
      float h = c[v] + bias;
      h = h > 0.f ? h : 0.f;               // ReLU
      p0[v] = fmaf(h, w2c0, p0[v]);
      p1[v] = fmaf(h, w2c1, p1[v]);
    }
  }

  // reduce the 16 column-lanes of each half (wave32: masks 8..1 stay in-half)
#pragma unroll
  for (int v = 0; v < 8; ++v) {
#pragma unroll
    for (int m = 8; m >= 1; m >>= 1) {
      p0[v] += __shfl_xor(p0[v], m, 32);
      p1[v] += __shfl_xor(p1[v], m, 32);
    }
  }

  if (mrow == 0) {                         // lanes 0 and 16 own 8 rows each
#pragma unroll
    for (int v = 0; v < 8; ++v) {
      const long long m = r0 + half * 8 + v;
      float l0 = p0[v] + b2v0;
      float l1 = p1[v] + b2v1;
      // log(softmax + eps)
      float mx = fmaxf(l0, l1);
      float e0 = __expf(l0 - mx), e1 = __expf(l1 - mx);
      float inv = 1.f / (e0 + e1);
      float lg0 = __logf(e0 * inv + EPS_LOG);
      float lg1 = __logf(e1 * inv + EPS_LOG);
      // gumbel noise, TEMP == 1
      float u0 = uni[m * 2 + 0], u1 = uni[m * 2 + 1];
      float a0 = lg0 - __logf(-__logf(u0));
      float a1 = lg1 - __logf(-__logf(u1));
      float mx2 = fmaxf(a0, a1);
      float y0 = __expf(a0 - mx2);
      float y1 = __expf(a1 - mx2);
      float ps = 1.f - y0 / (y0 + y1);
      ps = fminf(fmaxf(ps, 1e-6f), 1.f);
      poss[m] = ps;
      atomicAdd(&hist1[__float_as_uint(ps) >> 15], 1u);
    }
  }
}

// ------------------------------------------------ level-1 radix select
__global__ void __launch_bounds__(1024)
scan1_kernel(const unsigned int* __restrict__ hist1, unsigned int* __restrict__ scal)
{
  __shared__ unsigned int psum[1024];
  const int tid = threadIdx.x;
  const int CH = NBINS1 / 1024;            // 128
  unsigned int s = 0;
  for (int i = 0; i < CH; ++i) s += hist1[tid * CH + i];
  psum[tid] = s;
  __syncthreads();
  if (tid == 0) {
    unsigned int run = 0;
    for (int i = 0; i < 1024; ++i) { unsigned int t = psum[i]; psum[i] = run; run += t; }
  }
  __syncthreads();
  unsigned int c = psum[tid];
  for (int i = 0; i < CH; ++i) {
    unsigned int h = hist1[tid * CH + i];
    if (c < NUM_DROP && c + h >= NUM_DROP) {
      scal[0] = (unsigned int)(tid * CH + i);  // threshold bin (top 17 bits)
      scal[1] = c;                             // count strictly below it
    }
    c += h;
  }
}

// ---------------------------------- level-2 histogram inside threshold bin
__global__ void __launch_bounds__(256)
subhist_kernel(const float* __restrict__ poss,
               const unsigned int* __restrict__ scal,
               unsigned int* __restrict__ hist2)
{
  int i = blockIdx.x * blockDim.x + threadIdx.x;
  if (i >= M_TOT) return;
  unsigned int key = __float_as_uint(poss[i]);
  if ((key >> 15) == scal[0]) atomicAdd(&hist2[key & 0x7FFFu], 1u);
}

__global__ void __launch_bounds__(1024)
scan2_kernel(const unsigned int* __restrict__ hist2, unsigned int* __restrict__ scal)
{
  __shared__ unsigned int psum[1024];
  const int tid = threadIdx.x;
  const int CH = NBINS2 / 1024;            // 32
  unsigned int s = 0;
  for (int i = 0; i < CH; ++i) s += hist2[tid * CH + i];
  psum[tid] = s;
  __syncthreads();
  if (tid == 0) {
    unsigned int run = scal[1];
    for (int i = 0; i < 1024; ++i) { unsigned int t = psum[i]; psum[i] = run; run += t; }
  }
  __syncthreads();
  unsigned int c = psum[tid];
  for (int i = 0; i < CH; ++i) {
    unsigned int h = hist2[tid * CH + i];
    if (c < NUM_DROP && c + h >= NUM_DROP)
      scal[2] = (scal[0] << 15) | (unsigned int)(tid * CH + i); // exact fp32 bits of k-th value
    c += h;
  }
}

// ------------------------------------------------------- apply drop mask
__global__ void __launch_bounds__(256)
apply_drop_kernel(const float* __restrict__ x,
                  const float* __restrict__ poss,
                  const unsigned int* __restrict__ scal,
                  float* __restrict__ out)
{
  int i = blockIdx.x * blockDim.x + threadIdx.x;
  if (i >= M_TOT) return;
  float p  = poss[i];
  float xi = x[i];
  out[i] = (__float_as_uint(p) <= scal[2]) ? xi * p : xi;
}

// ------------------------------------------------------------------ entry
extern "C" void kernel_launch(void* const* d_in, const int* in_sizes, int n_in,
                              void* d_out, int out_size, void* d_ws, size_t ws_size,
                              hipStream_t stream) {
  const float* x    = (const float*)d_in[0];
  const void*  eidx = d_in[1];                 // int64 [2, N_EDGES], pass-through
  const float* ew   = (const float*)d_in[2];   // pass-through
  const float* eig  = (const float*)d_in[3];
  const float* uni  = (const float*)d_in[4];
  const float* W1   = (const float*)d_in[5];
  const float* b1   = (const float*)d_in[6];
  const float* W2   = (const float*)d_in[7];
  const float* b2   = (const float*)d_in[8];

  float*        poss  = (float*)d_ws;
  unsigned int* hist1 = (unsigned int*)((char*)d_ws + (size_t)M_TOT * 4);
  unsigned int* hist2 = hist1 + NBINS1;
  unsigned int* scal  = hist2 + NBINS2;        // [0]=bin, [1]=below, [2]=threshold key

  zero_ws_kernel<<<(NBINS1 + NBINS2 + 4 + 255) / 256, 256, 0, stream>>>(hist1, NBINS1 + NBINS2 + 4);
  gemm_poss_kernel<<<M_TOT / 128, 256, 0, stream>>>(eig, W1, b1, W2, b2, uni, poss, hist1);
  scan1_kernel<<<1, 1024, 0, stream>>>(hist1, scal);
  subhist_kernel<<<(M_TOT + 255) / 256, 256, 0, stream>>>(poss, scal, hist2);
  scan2_kernel<<<1, 1024, 0, stream>>>(hist2, scal);
  apply_drop_kernel<<<(M_TOT + 255) / 256, 256, 0, stream>>>(x, poss, scal, (float*)d_out);

  // tuple pass-through: x_new | edge_index (int64) | edge_weights (f32)
  char* outb = (char*)d_out + (size_t)M_TOT * sizeof(float);
  hipMemcpyAsync(outb, eidx, (size_t)2 * N_EDGES * sizeof(long long),
                 hipMemcpyDeviceToDevice, stream);
  hipMemcpyAsync(outb + (size_t)2 * N_EDGES * sizeof(long long), ew,
                 (size_t)N_EDGES * sizeof(float), hipMemcpyDeviceToDevice, stream);
}